// HistogramObserver_32521492365329
// MI455X (gfx1250) — compile-verified
//
#include <hip/hip_runtime.h>

// HistogramObserver forward for MI455X (gfx1250, wave32).
// Bandwidth-bound: 2 streaming passes over 256 MB -> ~22us at 23.3 TB/s.
// Pass 1: min/max via wave32 shuffles + u32-monotone-mapped global atomics.
// Pass 2: 2048-bin histogram via per-wave LDS-privatized ds_add_u32 atomics
//         (8 copies, 66 KB LDS -- exploits CDNA5's 320 KB/WGP LDS).
// Finalize: re-bin running_hist into the new grid (ds_add_f32 in LDS).

#define BINS 2048
#define NCOPIES 8      // one LDS histogram copy per wave (256 threads = 8 waves)
#define CSTRIDE 2064   // 2048 + 16 dwords pad -> copies start on different banks

// Monotone increasing map: float order -> unsigned order (no NaNs in this data).
__device__ __forceinline__ unsigned int fmap(float f) {
    unsigned int u = __float_as_uint(f);
    return (u & 0x80000000u) ? ~u : (u | 0x80000000u);
}
__device__ __forceinline__ float funmap(unsigned int m) {
    return __uint_as_float((m & 0x80000000u) ? (m ^ 0x80000000u) : ~m);
}

// ws layout (u32): [0]=mapped min, [1]=mapped max, [2..2+BINS)=u32 hist accum
__global__ void ho_init_kernel(const float* __restrict__ rmin,
                               const float* __restrict__ rmax,
                               unsigned int* __restrict__ ws) {
    int t = threadIdx.x;
    if (t == 0) ws[0] = fmap(rmin[0]);
    if (t == 1) ws[1] = fmap(rmax[0]);
    for (int i = t; i < BINS; i += blockDim.x) ws[2 + i] = 0u;
}

__global__ void ho_minmax_kernel(const float* __restrict__ xf, long long n,
                                 unsigned int* __restrict__ ws) {
    const float4* __restrict__ xv = (const float4*)xf;
    long long nvec = n >> 2;
    long long stride = (long long)gridDim.x * blockDim.x;
    long long gtid = (long long)blockIdx.x * blockDim.x + threadIdx.x;

    float vmin = INFINITY, vmax = -INFINITY;
    for (long long i = gtid; i < nvec; i += stride) {
        float4 v = xv[i];                       // global_load_b128
        vmin = fminf(vmin, fminf(fminf(v.x, v.y), fminf(v.z, v.w)));
        vmax = fmaxf(vmax, fmaxf(fmaxf(v.x, v.y), fmaxf(v.z, v.w)));
    }
    if (gtid < (n & 3LL)) {                     // scalar tail (n%4 elements)
        float v = xf[(nvec << 2) + gtid];
        vmin = fminf(vmin, v);
        vmax = fmaxf(vmax, v);
    }
    // wave32 butterfly reduction
    #pragma unroll
    for (int m = 16; m >= 1; m >>= 1) {
        vmin = fminf(vmin, __shfl_xor(vmin, m, 32));
        vmax = fmaxf(vmax, __shfl_xor(vmax, m, 32));
    }
    __shared__ float smin[8], smax[8];
    int wave = threadIdx.x >> 5;
    int lane = threadIdx.x & 31;
    if (lane == 0) { smin[wave] = vmin; smax[wave] = vmax; }
    __syncthreads();
    if (threadIdx.x == 0) {
        int nw = blockDim.x >> 5;
        float bmin = smin[0], bmax = smax[0];
        for (int w = 1; w < nw; ++w) {
            bmin = fminf(bmin, smin[w]);
            bmax = fmaxf(bmax, smax[w]);
        }
        atomicMin(&ws[0], fmap(bmin));          // global_atomic_min_u32
        atomicMax(&ws[1], fmap(bmax));          // global_atomic_max_u32
    }
}

__global__ void ho_hist_kernel(const float* __restrict__ xf, long long n,
                               unsigned int* __restrict__ ws) {
    __shared__ unsigned int lh[NCOPIES * CSTRIDE];   // ~66 KB LDS (320 KB/WGP on CDNA5)
    for (int i = threadIdx.x; i < NCOPIES * CSTRIDE; i += blockDim.x) lh[i] = 0u;

    float new_min = funmap(ws[0]);
    float new_max = funmap(ws[1]);
    float w = (new_max - new_min) / (float)BINS;
    float safe_w = (w > 0.0f) ? w : 1.0f;
    __syncthreads();

    // one private copy per wave -> intra-wave conflicts only
    unsigned int* __restrict__ myh = &lh[(threadIdx.x >> 5) * CSTRIDE];

    const float4* __restrict__ xv = (const float4*)xf;
    long long nvec = n >> 2;
    long long stride = (long long)gridDim.x * blockDim.x;
    long long gtid = (long long)blockIdx.x * blockDim.x + threadIdx.x;
    for (long long i = gtid; i < nvec; i += stride) {
        float4 v = xv[i];
        // exact f32 division to match reference bin-edge rounding
        int i0 = min(max((int)floorf((v.x - new_min) / safe_w), 0), BINS - 1);
        int i1 = min(max((int)floorf((v.y - new_min) / safe_w), 0), BINS - 1);
        int i2 = min(max((int)floorf((v.z - new_min) / safe_w), 0), BINS - 1);
        int i3 = min(max((int)floorf((v.w - new_min) / safe_w), 0), BINS - 1);
        atomicAdd(&myh[i0], 1u);                // ds_add_u32
        atomicAdd(&myh[i1], 1u);
        atomicAdd(&myh[i2], 1u);
        atomicAdd(&myh[i3], 1u);
    }
    if (gtid < (n & 3LL)) {
        float v = xf[(nvec << 2) + gtid];
        int i0 = min(max((int)floorf((v - new_min) / safe_w), 0), BINS - 1);
        atomicAdd(&myh[i0], 1u);
    }
    __syncthreads();

    // flush privatized copies -> global u32 accumulator (deterministic int adds)
    for (int b = threadIdx.x; b < BINS; b += blockDim.x) {
        unsigned int s = 0;
        #pragma unroll
        for (int c = 0; c < NCOPIES; ++c) s += lh[c * CSTRIDE + b];
        if (s) atomicAdd(&ws[2 + b], s);        // global_atomic_add_u32
    }
}

__global__ void ho_finalize_kernel(const float* __restrict__ running_hist,
                                   const float* __restrict__ rmin_p,
                                   const float* __restrict__ rmax_p,
                                   const unsigned int* __restrict__ ws,
                                   float* __restrict__ out) {
    __shared__ float sh[BINS];
    int t = threadIdx.x;
    for (int i = t; i < BINS; i += blockDim.x) sh[i] = (float)ws[2 + i];
    __syncthreads();

    float new_min = funmap(ws[0]);
    float new_max = funmap(ws[1]);
    float r_min = rmin_p[0], r_max = rmax_p[0];

    float dst_w = (new_max - new_min) / (float)BINS;
    float safe_w = (dst_w > 0.0f) ? dst_w : 1.0f;
    float src_w = (r_max - r_min) / (float)BINS;
    float safe_sw = (src_w > 0.0f) ? src_w : 1.0f;
    bool degen = (src_w == 0.0f) || (dst_w == 0.0f);

    for (int i = t; i < BINS; i += blockDim.x) {
        float fi = (float)i;
        float src_begin = r_min + src_w * fi;
        float src_end = src_begin + src_w;
        int db1 = min(max((int)floorf((src_begin - new_min) / safe_w), 0), BINS - 1);
        int db2 = min(max((int)floorf((src_end   - new_min) / safe_w), 0), BINS - 1);
        float dst_bin_end = new_min + dst_w * ((float)db1 + 1.0f);
        float cnt = running_hist[i];
        // jnp.round == round-half-even == rintf (default RNE)
        float frac = fminf(rintf((dst_bin_end - src_begin) / safe_sw * cnt), cnt);
        float d1 = degen ? cnt : frac;
        atomicAdd(&sh[db1], d1);                // ds_add_f32
        atomicAdd(&sh[db2], cnt - d1);
    }
    __syncthreads();

    for (int i = t; i < BINS; i += blockDim.x) out[i] = sh[i];
    if (t == 0) { out[BINS] = new_min; out[BINS + 1] = new_max; }
}

extern "C" void kernel_launch(void* const* d_in, const int* in_sizes, int n_in,
                              void* d_out, int out_size, void* d_ws, size_t ws_size,
                              hipStream_t stream) {
    const float* x     = (const float*)d_in[0];
    const float* rhist = (const float*)d_in[1];
    const float* rmin  = (const float*)d_in[2];
    const float* rmax  = (const float*)d_in[3];
    float* out = (float*)d_out;
    unsigned int* ws = (unsigned int*)d_ws;

    long long n = (long long)in_sizes[0];
    long long nvec = n >> 2;
    const int threads = 256;
    long long want = (nvec + threads - 1) / threads;

    int blocksB = (int)(want < 8192 ? (want < 1 ? 1 : want) : 8192);
    int blocksC = (int)(want < 8192 ? (want < 1 ? 1 : want) : 8192);

    ho_init_kernel<<<1, 256, 0, stream>>>(rmin, rmax, ws);
    ho_minmax_kernel<<<blocksB, threads, 0, stream>>>(x, n, ws);
    ho_hist_kernel<<<blocksC, threads, 0, stream>>>(x, n, ws);
    ho_finalize_kernel<<<1, 1024, 0, stream>>>(rhist, rmin, rmax, ws, out);
}